// SigLoss_51393578664084
// MI455X (gfx1250) — compile-verified
//
#include <hip/hip_runtime.h>

typedef float v2f __attribute__((ext_vector_type(2)));
typedef float v8f __attribute__((ext_vector_type(8)));

#define LSEQ   512   // padded column count (511 real increments + 1 zero pad)
#define LINC   511   // number of increments (rows/cols of inc)
#define STR    17    // padded LDS row stride (floats) for dX/dY -> conflict-free
#define BLOCK  512   // 16 wave32s

__global__ void sigloss_init_kernel(float* out) { out[0] = 0.0f; }

__global__ __launch_bounds__(BLOCK)
void sigloss_pde_kernel(const float* __restrict__ X,
                        const float* __restrict__ Y,
                        float* __restrict__ out) {
    extern __shared__ float smem[];
    float* dXs   = smem;                    // 512*17 floats
    float* dYs   = dXs + LSEQ * STR;        // 512*17 floats
    float* strip = dYs + LSEQ * STR;        // 16*512  floats: inc strip (16 rows)
    float* krow  = strip + 16 * LSEQ;       // 512     floats: live K row
    float* wsum  = krow + LSEQ;             // 16      floats: per-wave scan sums

    const int t    = threadIdx.x;
    const int lane = t & 31;
    const int wave = t >> 5;

    const int p    = blockIdx.x >> 7;       // 0: k(X,X)  1: k(Y,Y)  2: k(X,Y)
    const int bidx = blockIdx.x & 127;      // batch element

    const float* Aseq = (p == 1) ? Y : X;
    const float* Bseq = (p == 0) ? X : Y;
    const float  wgt  = (p == 2) ? -2.0f : 1.0f;

    // ---- load increments dA, dB into LDS (thread t owns time-step t) ----
    {
        const size_t base = (size_t)bidx * 512 * 16;
        if (t < LINC) {
            const float4* a0 = reinterpret_cast<const float4*>(Aseq + base + (size_t)t * 16);
            const float4* b0 = reinterpret_cast<const float4*>(Bseq + base + (size_t)t * 16);
#pragma unroll
            for (int q = 0; q < 4; ++q) {
                float4 xa = a0[q], xb = a0[q + 4];   // rows t and t+1
                float4 ya = b0[q], yb = b0[q + 4];
                dXs[t * STR + 4 * q + 0] = xb.x - xa.x;
                dXs[t * STR + 4 * q + 1] = xb.y - xa.y;
                dXs[t * STR + 4 * q + 2] = xb.z - xa.z;
                dXs[t * STR + 4 * q + 3] = xb.w - xa.w;
                dYs[t * STR + 4 * q + 0] = yb.x - ya.x;
                dYs[t * STR + 4 * q + 1] = yb.y - ya.y;
                dYs[t * STR + 4 * q + 2] = yb.z - ya.z;
                dYs[t * STR + 4 * q + 3] = yb.w - ya.w;
            }
        } else {
#pragma unroll
            for (int d = 0; d < 16; ++d) {
                dXs[t * STR + d] = 0.0f;
                dYs[t * STR + d] = 0.0f;
            }
        }
        krow[t] = 1.0f;                      // K[0, :] = 1
    }
    __syncthreads();

    const int m  = lane & 15;
    const int hi = lane >> 4;                // 0 for lanes 0-15, 1 for 16-31

    // ---- sweep 32 row-blocks of 16 rows each ----
    for (int ib = 0; ib < 32; ++ib) {
        const int i0 = ib * 16;

        // (1) WMMA: strip[r][j] = <dX[i0+r], dY[j]> - 1  for 16 rows x 512 cols.
        //     Each wave owns 2 column tiles of 16; K=16 folded as 4x k-steps of
        //     V_WMMA_F32_16X16X4_F32.
#pragma unroll
        for (int tt = 0; tt < 2; ++tt) {
            const int jt = wave * 2 + tt;    // column tile 0..31
            v8f c = {};
#pragma unroll
            for (int kk = 0; kk < 16; kk += 4) {
                const int ka = kk + 2 * hi;  // ISA 16x4 f32 A/B fragment layout
                v2f av, bv;
                av.x = dXs[(i0 + m) * STR + ka];
                av.y = dXs[(i0 + m) * STR + ka + 1];
                bv.x = dYs[(jt * 16 + m) * STR + ka];
                bv.y = dYs[(jt * 16 + m) * STR + ka + 1];
                c = __builtin_amdgcn_wmma_f32_16x16x4_f32(
                        false, av, false, bv, (short)0, c, false, false);
            }
#pragma unroll
            for (int v = 0; v < 8; ++v) {    // C layout: M = v + 8*hi, N = m
                strip[(v + 8 * hi) * LSEQ + jt * 16 + m] = c[v] - 1.0f;
            }
        }
        __syncthreads();

        // (2) recurrence over the 16 rows of this strip:
        //     b[j] = K[i-1,j+1] + K[i-1,j]*inc[i-1,j];  K[i,j+1] = 1 + incl_scan(b)[j]
        const int rmax = (i0 + 16 <= LINC) ? 16 : (LINC - i0);
        for (int r = 0; r < rmax; ++r) {
            float bj = 0.0f;
            if (t < LINC) {
                bj = krow[t + 1] + krow[t] * strip[r * LSEQ + t];
            }
            // wave32 inclusive scan
            float s = bj;
#pragma unroll
            for (int d = 1; d < 32; d <<= 1) {
                float u = __shfl_up(s, d, 32);
                if (lane >= d) s += u;
            }
            if (lane == 31) wsum[wave] = s;
            __syncthreads();
            float off = 0.0f;
            for (int u = 0; u < wave; ++u) off += wsum[u];
            if (t < LINC) krow[t + 1] = 1.0f + s + off;
            __syncthreads();
        }
    }

    if (t == 0) {
        atomicAdd(out, wgt * krow[LINC] * (1.0f / 128.0f));
    }
}

extern "C" void kernel_launch(void* const* d_in, const int* in_sizes, int n_in,
                              void* d_out, int out_size, void* d_ws, size_t ws_size,
                              hipStream_t stream) {
    (void)in_sizes; (void)n_in; (void)out_size; (void)d_ws; (void)ws_size;
    const float* X = (const float*)d_in[0];
    const float* Y = (const float*)d_in[1];
    float* out = (float*)d_out;

    sigloss_init_kernel<<<1, 1, 0, stream>>>(out);

    const size_t shmem =
        (size_t)(2 * LSEQ * STR + 16 * LSEQ + LSEQ + 16) * sizeof(float);
    // 384 workgroups: 3 pair types x 128 batch elements
    sigloss_pde_kernel<<<384, BLOCK, shmem, stream>>>(X, Y, out);
}